// BLOCK_APPNP_89859305766970
// MI455X (gfx1250) — compile-verified
//
#include <hip/hip_runtime.h>
#include <cstddef>

#define N_NODES  100000
#define N_EDGES  1600000
#define F_IN     128
#define F_HID    64
#define F_OUT    40
#define K_ITERS  10
#define ALPHA    0.1f

typedef __attribute__((ext_vector_type(2))) float v2f;
typedef __attribute__((ext_vector_type(8))) float v8f;

// ---------------------------------------------------------------------------
// GCN normalization: deg (with self loop), dinv = rsqrt(deg), norm per edge
// ---------------------------------------------------------------------------
__global__ void deg_init_kernel(float* __restrict__ deg, int n) {
    int i = blockIdx.x * blockDim.x + threadIdx.x;
    if (i < n) deg[i] = 1.0f;   // self loop contributes 1
}

__global__ void deg_count_kernel(const int* __restrict__ dst, float* __restrict__ deg, int e) {
    int i = blockIdx.x * blockDim.x + threadIdx.x;
    if (i < e) atomicAdd(&deg[dst[i]], 1.0f);
}

__global__ void dinv_kernel(const float* __restrict__ deg, float* __restrict__ dinv, int n) {
    int i = blockIdx.x * blockDim.x + threadIdx.x;
    if (i < n) dinv[i] = rsqrtf(deg[i]);   // deg >= 1 always (self loop)
}

__global__ void norm_kernel(const int* __restrict__ src, const int* __restrict__ dst,
                            const float* __restrict__ dinv, float* __restrict__ nrm, int e) {
    int i = blockIdx.x * blockDim.x + threadIdx.x;
    if (i < e) nrm[i] = dinv[src[i]] * dinv[dst[i]];
}

// ---------------------------------------------------------------------------
// GEMM1: H[N,64] = relu(X[N,128] @ W[128,64] + b)  via V_WMMA_F32_16X16X4_F32
// One wave handles a 16-row strip x all 4 column tiles (4 accumulators).
// 2 waves / block, grid = 6250/2 = 3125 (N = 6250*16 exactly).
// ---------------------------------------------------------------------------
__global__ __launch_bounds__(64) void gemm1_relu_kernel(
    const float* __restrict__ X, const float* __restrict__ W,
    const float* __restrict__ bias, float* __restrict__ H) {
    const int wave = threadIdx.x >> 5;
    const int lane = threadIdx.x & 31;
    const int row0 = (blockIdx.x * 2 + wave) * 16;
    const int mr   = lane & 15;    // row (A) / col (B,C,D) within tile
    const int hi   = lane >> 4;    // lane-half selector
    const int koff = hi * 2;       // K offset for A/B fragments

    v8f acc[4] = {};
    const float* __restrict__ xrow = X + (size_t)(row0 + mr) * F_IN;

    #pragma unroll 4
    for (int k0 = 0; k0 < F_IN; k0 += 4) {
        v2f a;
        a.x = xrow[k0 + koff];
        a.y = xrow[k0 + koff + 1];
        #pragma unroll
        for (int t = 0; t < 4; ++t) {
            v2f bf;
            bf.x = W[(size_t)(k0 + koff) * F_HID + t * 16 + mr];
            bf.y = W[(size_t)(k0 + koff + 1) * F_HID + t * 16 + mr];
            acc[t] = __builtin_amdgcn_wmma_f32_16x16x4_f32(
                false, a, false, bf, (short)0, acc[t], false, false);
        }
    }

    #pragma unroll
    for (int t = 0; t < 4; ++t) {
        const float bv = bias[t * 16 + mr];
        #pragma unroll
        for (int i = 0; i < 8; ++i) {
            const int m = (hi << 3) + i;
            float v = acc[t][i] + bv;
            H[(size_t)(row0 + m) * F_HID + t * 16 + mr] = v > 0.0f ? v : 0.0f;
        }
    }
}

// ---------------------------------------------------------------------------
// GEMM2: H[N,40] = relu(P[N,64] @ W[64,40] + b), cols padded 40 -> 48 (3 tiles)
// ---------------------------------------------------------------------------
__global__ __launch_bounds__(64) void gemm2_relu_kernel(
    const float* __restrict__ P, const float* __restrict__ W,
    const float* __restrict__ bias, float* __restrict__ H) {
    const int wave = threadIdx.x >> 5;
    const int lane = threadIdx.x & 31;
    const int row0 = (blockIdx.x * 2 + wave) * 16;
    const int mr   = lane & 15;
    const int hi   = lane >> 4;
    const int koff = hi * 2;

    v8f acc[3] = {};
    const float* __restrict__ prow = P + (size_t)(row0 + mr) * F_HID;

    #pragma unroll 4
    for (int k0 = 0; k0 < F_HID; k0 += 4) {
        v2f a;
        a.x = prow[k0 + koff];
        a.y = prow[k0 + koff + 1];
        #pragma unroll
        for (int t = 0; t < 3; ++t) {
            const int c = t * 16 + mr;
            v2f bf;
            bf.x = (c < F_OUT) ? W[(size_t)(k0 + koff) * F_OUT + c] : 0.0f;
            bf.y = (c < F_OUT) ? W[(size_t)(k0 + koff + 1) * F_OUT + c] : 0.0f;
            acc[t] = __builtin_amdgcn_wmma_f32_16x16x4_f32(
                false, a, false, bf, (short)0, acc[t], false, false);
        }
    }

    #pragma unroll
    for (int t = 0; t < 3; ++t) {
        const int c = t * 16 + mr;
        if (c < F_OUT) {
            const float bv = bias[c];
            #pragma unroll
            for (int i = 0; i < 8; ++i) {
                const int m = (hi << 3) + i;
                float v = acc[t][i] + bv;
                H[(size_t)(row0 + m) * F_OUT + c] = v > 0.0f ? v : 0.0f;
            }
        }
    }
}

// ---------------------------------------------------------------------------
// APPNP step, part 1: out = ALPHA*h + (1-ALPHA)*dinv[i]^2 * xc  (teleport +
// self-loop edge folded in; also fully initializes `out` before the scatter)
// ---------------------------------------------------------------------------
template <int F>
__global__ void appnp_init_kernel(const float* __restrict__ h, const float* __restrict__ xc,
                                  const float* __restrict__ dinv, float* __restrict__ out,
                                  int total) {
    int t = blockIdx.x * blockDim.x + threadIdx.x;
    if (t >= total) return;
    const int i = t / F;
    const float dn = dinv[i];
    out[t] = ALPHA * h[t] + (1.0f - ALPHA) * dn * dn * xc[t];
}

// ---------------------------------------------------------------------------
// APPNP step, part 2: edge-parallel scatter
// out[dst,f] += (1-ALPHA)*norm[e] * xc[src,f];  64 feature lanes per edge
// (coalesced row gather + contiguous atomic row scatter), 4 edges per block.
// ---------------------------------------------------------------------------
template <int F>
__global__ __launch_bounds__(256) void appnp_scatter_kernel(
    const float* __restrict__ xc, const int* __restrict__ src, const int* __restrict__ dst,
    const float* __restrict__ nrm, float* __restrict__ out, int e) {
    const int eid = blockIdx.x * 4 + threadIdx.y;
    if (eid >= e) return;
    const int f = threadIdx.x;
    if (f >= F) return;
    const int s = src[eid];
    const int d = dst[eid];
    const float w = (1.0f - ALPHA) * nrm[eid];
    atomicAdd(&out[(size_t)d * F + f], w * xc[(size_t)s * F + f]);
}

// ---------------------------------------------------------------------------
// log_softmax over 40 classes, one thread per node
// ---------------------------------------------------------------------------
__global__ void log_softmax_kernel(const float* __restrict__ in, float* __restrict__ out, int n) {
    int i = blockIdx.x * blockDim.x + threadIdx.x;
    if (i >= n) return;
    const float* __restrict__ r = in + (size_t)i * F_OUT;
    float mx = r[0];
    #pragma unroll
    for (int f = 1; f < F_OUT; ++f) mx = fmaxf(mx, r[f]);
    float s = 0.0f;
    #pragma unroll
    for (int f = 0; f < F_OUT; ++f) s += expf(r[f] - mx);
    const float ls = logf(s);
    float* __restrict__ o = out + (size_t)i * F_OUT;
    #pragma unroll
    for (int f = 0; f < F_OUT; ++f) o[f] = r[f] - mx - ls;
}

// ---------------------------------------------------------------------------
// host-side launcher
// ---------------------------------------------------------------------------
extern "C" void kernel_launch(void* const* d_in, const int* in_sizes, int n_in,
                              void* d_out, int out_size, void* d_ws, size_t ws_size,
                              hipStream_t stream) {
    (void)in_sizes; (void)n_in; (void)out_size; (void)ws_size;

    const float* x  = (const float*)d_in[0];
    const int*   ei = (const int*)d_in[1];
    const float* W1 = (const float*)d_in[2];
    const float* b1 = (const float*)d_in[3];
    const float* W2 = (const float*)d_in[4];
    const float* b2 = (const float*)d_in[5];
    float* out = (float*)d_out;

    const int* src = ei;             // edge_index[0]
    const int* dst = ei + N_EDGES;   // edge_index[1]

    // workspace layout (all fp32, fully written before read)
    float* ws   = (float*)d_ws;
    float* deg  = ws;                               // N
    float* dinv = deg  + N_NODES;                   // N
    float* nrm  = dinv + N_NODES;                   // E
    float* h1   = nrm  + N_EDGES;                   // N*64 (teleport L1; later h2 @ stride 40)
    float* bufA = h1   + (size_t)N_NODES * F_HID;   // N*64 ping
    float* bufB = bufA + (size_t)N_NODES * F_HID;   // N*64 pong

    // --- GCN symmetric normalization -------------------------------------
    deg_init_kernel<<<(N_NODES + 255) / 256, 256, 0, stream>>>(deg, N_NODES);
    deg_count_kernel<<<(N_EDGES + 255) / 256, 256, 0, stream>>>(dst, deg, N_EDGES);
    dinv_kernel<<<(N_NODES + 255) / 256, 256, 0, stream>>>(deg, dinv, N_NODES);
    norm_kernel<<<(N_EDGES + 255) / 256, 256, 0, stream>>>(src, dst, dinv, nrm, N_EDGES);

    const int rowBlocks = (N_NODES / 16) / 2;   // 3125 (exact: N = 6250*16)
    const dim3 sblk(64, 4);
    const int sgrid = (N_EDGES + 3) / 4;

    // --- layer 1: Linear + ReLU, then K APPNP steps at F=64 ---------------
    gemm1_relu_kernel<<<rowBlocks, 64, 0, stream>>>(x, W1, b1, h1);
    {
        const float* xc = h1;
        float* pp[2] = {bufA, bufB};
        for (int k = 0; k < K_ITERS; ++k) {
            float* o = pp[k & 1];
            appnp_init_kernel<F_HID><<<(N_NODES * F_HID + 255) / 256, 256, 0, stream>>>(
                h1, xc, dinv, o, N_NODES * F_HID);
            appnp_scatter_kernel<F_HID><<<sgrid, sblk, 0, stream>>>(
                xc, src, dst, nrm, o, N_EDGES);
            xc = o;
        }
        // K_ITERS even -> result in bufB
    }

    // --- layer 2: Linear + ReLU (reuse h1 buffer, stride 40), K steps -----
    gemm2_relu_kernel<<<rowBlocks, 64, 0, stream>>>(bufB, W2, b2, h1);
    const float* xc2 = h1;
    {
        float* pp[2] = {bufA, bufB};
        for (int k = 0; k < K_ITERS; ++k) {
            float* o = pp[k & 1];
            appnp_init_kernel<F_OUT><<<(N_NODES * F_OUT + 255) / 256, 256, 0, stream>>>(
                h1, xc2, dinv, o, N_NODES * F_OUT);
            appnp_scatter_kernel<F_OUT><<<sgrid, sblk, 0, stream>>>(
                xc2, src, dst, nrm, o, N_EDGES);
            xc2 = o;
        }
        // result in bufB
    }

    // --- log_softmax -------------------------------------------------------
    log_softmax_kernel<<<(N_NODES + 255) / 256, 256, 0, stream>>>(bufB, out, N_NODES);
}